// GCN_20933670600928
// MI455X (gfx1250) — compile-verified
//
#include <hip/hip_runtime.h>
#include <math.h>

// ---------------------------------------------------------------------------
// 2-layer GCN for MI455X (gfx1250, wave32).
//   h1   = x @ W1                       (WMMA f32 16x16x4 GEMM, N x 256 x 30)
//   agg1 = Dhat^-1/2 (A+I) Dhat^-1/2 h1 (f32 atomics; L2-resident accumulator)
//   hr   = relu(agg1 + b1); h2 = hr @ W2 (fused per-node)
//   out  = sigmoid(agg(h2) + b2)
// HBM floor ~150 MB/call (x once, edge list twice) => ~7 us at 23.3 TB/s.
// ---------------------------------------------------------------------------

typedef __attribute__((ext_vector_type(2))) float v2f;
typedef __attribute__((ext_vector_type(8))) float v8f;

#define D_FEAT   256
#define D_HID    30
#define D_HIDP   32     // padded hidden dim (two 16-col WMMA tiles)

// ---------------------------------------------------------------------------
// 0) init: deg=1 (self loop), zero agg buffers
// ---------------------------------------------------------------------------
__global__ void gcn_init(float* __restrict__ deg, float* __restrict__ agg1,
                         float* __restrict__ agg2, int N) {
    long t = (long)blockIdx.x * blockDim.x + threadIdx.x;
    long total = (long)N * D_HIDP;
    if (t < total) agg1[t] = 0.0f;
    if (t < N) { deg[t] = 1.0f; agg2[t] = 0.0f; }
}

// ---------------------------------------------------------------------------
// 1) in-degree accumulation at target (col)
// ---------------------------------------------------------------------------
__global__ void gcn_degree(const int* __restrict__ colv, float* __restrict__ deg, int E) {
    int e = blockIdx.x * blockDim.x + threadIdx.x;
    if (e >= E) return;
    atomicAdd(&deg[colv[e]], 1.0f);
}

// deg -> rsqrt(deg) in place (deg >= 1 always)
__global__ void gcn_rsqrt(float* __restrict__ deg, int N) {
    int n = blockIdx.x * blockDim.x + threadIdx.x;
    if (n >= N) return;
    deg[n] = rsqrtf(deg[n]);
}

// ---------------------------------------------------------------------------
// 2) h1 = x @ W1  via V_WMMA_F32_16X16X4_F32.
//    One wave computes a 16-row x 32-col tile (2 accumulators), K looped by 4.
//    A 16x4 layout: lanes 0-15 hold rows M=0..15; VGPR0/1 = K={0,1} for the
//    low lane half, K={2,3} for the high half. B 4x16 mirrors this with N on
//    lanes. W1 is staged in LDS pre-paired in fragment order so each lane's
//    B operand is a single aligned ds_load_b64. Out-of-range A rows are
//    CLAMPED (branch-free b64 loads): row m of A only feeds row m of D, and
//    those D rows are never stored.
// ---------------------------------------------------------------------------
__global__ void __launch_bounds__(256)
gcn_gemm1(const float* __restrict__ x, const float* __restrict__ W1,
          float* __restrict__ h1, int N) {
    // BsP[p*32 + n] = (W1[2p][n], W1[2p+1][n]),  p = 0..127, n = 0..31 (padded)
    __shared__ float2 BsP[(D_FEAT / 2) * D_HIDP];   // 32 KB
    const int tid = threadIdx.x;
    for (int i = tid; i < (D_FEAT / 2) * D_HIDP; i += 256) {
        const int p = i >> 5, c = i & 31;
        float2 v;
        v.x = (c < D_HID) ? W1[(2 * p)     * D_HID + c] : 0.0f;
        v.y = (c < D_HID) ? W1[(2 * p + 1) * D_HID + c] : 0.0f;
        BsP[i] = v;
    }
    __syncthreads();

    const int lane  = tid & 31;
    const int wave  = tid >> 5;
    const int tile  = blockIdx.x * 8 + wave;
    const long row0 = (long)tile * 16;
    if (row0 >= N) return;                  // wave-uniform: EXEC stays all-1s

    v8f acc0 = {};
    v8f acc1 = {};

    const int  m      = (int)row0 + (lane & 15);
    const int  mclamp = (m < N) ? m : (N - 1);     // branch-free guard
    const int  khalf  = (lane >> 4) << 1;   // 0 for lanes 0-15, 2 for 16-31
    const int  phalf  = lane >> 4;          // pair-index offset for this half
    const int  nlo    = lane & 15;
    const float*  arow = x + (size_t)mclamp * D_FEAT + khalf;
    const float2* bp0  = BsP + nlo;
    const float2* bp1  = BsP + 16 + nlo;

    for (int k = 0; k < D_FEAT; k += 4) {
        const float2 af = *(const float2*)(arow + k);        // global_load_b64
        const int    p  = (k >> 1) + phalf;
        const float2 bf0 = bp0[p << 5];                      // ds_load_b64
        const float2 bf1 = bp1[p << 5];                      // ds_load_b64
        v2f a;  a.x  = af.x;  a.y  = af.y;
        v2f b0; b0.x = bf0.x; b0.y = bf0.y;
        v2f b1; b1.x = bf1.x; b1.y = bf1.y;
        acc0 = __builtin_amdgcn_wmma_f32_16x16x4_f32(
            false, a, false, b0, (short)0, acc0, false, false);
        acc1 = __builtin_amdgcn_wmma_f32_16x16x4_f32(
            false, a, false, b1, (short)0, acc1, false, false);
    }

    // D layout: VGPR i -> M = i (+8 for high lane half), N = lane & 15
    const int mo = (lane >> 4) << 3;
    for (int i = 0; i < 8; ++i) {
        long r = row0 + mo + i;
        if (r < N) {
            h1[r * D_HIDP + nlo]      = acc0[i];
            h1[r * D_HIDP + 16 + nlo] = acc1[i];
        }
    }
}

// ---------------------------------------------------------------------------
// 3) layer-1 edge aggregation: 32 lanes per edge, lane d handles feature d.
//    Edge index loads broadcast within the wave; h1 row = one cacheline;
//    atomics coalesce into the L2-resident agg1.
// ---------------------------------------------------------------------------
__global__ void gcn_agg1(const int* __restrict__ rowv, const int* __restrict__ colv,
                         const float* __restrict__ h1, const float* __restrict__ dis,
                         float* __restrict__ agg1, long total) {
    long t = (long)blockIdx.x * blockDim.x + threadIdx.x;
    if (t >= total) return;
    const int e = (int)(t >> 5);
    const int d = (int)(t & 31);
    if (d >= D_HID) return;
    const int r = rowv[e];
    const int c = colv[e];
    const float nm = dis[r] * dis[c];
    atomicAdd(&agg1[(size_t)c * D_HIDP + d], h1[(size_t)r * D_HIDP + d] * nm);
}

// ---------------------------------------------------------------------------
// 4) fused: add self-loop term + bias, ReLU, dot with W2 -> h2
// ---------------------------------------------------------------------------
__global__ void gcn_relu_dot(const float* __restrict__ h1, const float* __restrict__ agg1,
                             const float* __restrict__ dis, const float* __restrict__ b1,
                             const float* __restrict__ W2, float* __restrict__ h2, int N) {
    int n = blockIdx.x * blockDim.x + threadIdx.x;
    if (n >= N) return;
    const float dn    = dis[n];
    const float selfn = dn * dn;
    const float* h1r  = h1   + (size_t)n * D_HIDP;
    const float* a1r  = agg1 + (size_t)n * D_HIDP;
    float acc = 0.0f;
#pragma unroll
    for (int d = 0; d < D_HID; ++d) {
        float v = a1r[d] + h1r[d] * selfn + b1[d];
        v = v > 0.0f ? v : 0.0f;
        acc += v * W2[d];
    }
    h2[n] = acc;
}

// ---------------------------------------------------------------------------
// 5) layer-2 edge aggregation: one scalar atomic per edge
// ---------------------------------------------------------------------------
__global__ void gcn_agg2(const int* __restrict__ rowv, const int* __restrict__ colv,
                         const float* __restrict__ h2, const float* __restrict__ dis,
                         float* __restrict__ agg2, int E) {
    int e = blockIdx.x * blockDim.x + threadIdx.x;
    if (e >= E) return;
    const int r = rowv[e];
    const int c = colv[e];
    atomicAdd(&agg2[c], h2[r] * dis[r] * dis[c]);
}

// ---------------------------------------------------------------------------
// 6) self-loop + bias + sigmoid -> d_out
// ---------------------------------------------------------------------------
__global__ void gcn_final(const float* __restrict__ agg2, const float* __restrict__ h2,
                          const float* __restrict__ dis, const float* __restrict__ b2,
                          float* __restrict__ out, int N) {
    int n = blockIdx.x * blockDim.x + threadIdx.x;
    if (n >= N) return;
    const float dn = dis[n];
    const float v  = agg2[n] + h2[n] * dn * dn + b2[0];
    out[n] = 1.0f / (1.0f + expf(-v));
}

// ---------------------------------------------------------------------------
extern "C" void kernel_launch(void* const* d_in, const int* in_sizes, int n_in,
                              void* d_out, int out_size, void* d_ws, size_t ws_size,
                              hipStream_t stream) {
    const float* x   = (const float*)d_in[0];
    const int*   ei  = (const int*)d_in[1];     // [2, E]
    const float* W1  = (const float*)d_in[2];   // [256, 30]
    const float* b1  = (const float*)d_in[3];   // [30]
    const float* W2  = (const float*)d_in[4];   // [30, 1]
    const float* b2  = (const float*)d_in[5];   // [1]

    const int N = out_size;                     // D_OUT == 1 -> out has N elems
    const int E = in_sizes[1] / 2;
    const int* rowv = ei;
    const int* colv = ei + E;

    // workspace layout (f32): dis | h1[N*32] | agg1[N*32] | h2[N] | agg2[N]
    float* ws   = (float*)d_ws;
    float* dis  = ws;
    float* h1   = dis  + (size_t)N;
    float* agg1 = h1   + (size_t)N * D_HIDP;
    float* h2   = agg1 + (size_t)N * D_HIDP;
    float* agg2 = h2   + (size_t)N;

    const int B = 256;
    const long initTot = (long)N * D_HIDP;
    gcn_init<<<(int)((initTot + B - 1) / B), B, 0, stream>>>(dis, agg1, agg2, N);
    gcn_degree<<<(E + B - 1) / B, B, 0, stream>>>(colv, dis, E);
    gcn_rsqrt<<<(N + B - 1) / B, B, 0, stream>>>(dis, N);

    const int tiles   = (N + 15) / 16;
    const int gblocks = (tiles + 7) / 8;        // 8 waves (tiles) per block
    gcn_gemm1<<<gblocks, B, 0, stream>>>(x, W1, h1, N);

    const long aggTot = (long)E * 32;
    gcn_agg1<<<(int)((aggTot + B - 1) / B), B, 0, stream>>>(rowv, colv, h1, dis, agg1, aggTot);
    gcn_relu_dot<<<(N + B - 1) / B, B, 0, stream>>>(h1, agg1, dis, b1, W2, h2, N);
    gcn_agg2<<<(E + B - 1) / B, B, 0, stream>>>(rowv, colv, h2, dis, agg2, E);
    gcn_final<<<(N + B - 1) / B, B, 0, stream>>>(agg2, h2, dis, b2, (float*)d_out, N);
}